// FFTConvModule_26774826123739
// MI455X (gfx1250) — compile-verified
//
#include <hip/hip_runtime.h>

// FFT-based causal convolution for MI455X (gfx1250), fp32 end-to-end.
// One workgroup (256 threads = 8 wave32) per (b,h) channel.
// FFT_4096 = 3 radix-16 stages; each stage = DFT16-matrix x (16x256) data,
// executed as complex 16x16 tile matmuls on V_WMMA_F32_16X16X4_F32.
// The x row is DMA'd into LDS by the Tensor Data Mover, overlapped with the
// filter FFT; completion enforced with s_wait_tensorcnt.

typedef float v2f __attribute__((ext_vector_type(2)));
typedef float v8f __attribute__((ext_vector_type(8)));
typedef unsigned int u32x4 __attribute__((ext_vector_type(4)));
typedef int i32x8 __attribute__((ext_vector_type(8)));
typedef int i32x4 __attribute__((ext_vector_type(4)));

#define PI2F 6.2831853071795864769f
#define NFFT 4096
#define NPAD 4352  // 4096 + 4096/16 padding (break stride-16 LDS bank conflicts)

__device__ __forceinline__ int slotf(int i) { return i + (i >> 4); }

__device__ __forceinline__ v8f wmma4(v2f a, v2f b, v8f c) {
  // (neg_a, A, neg_b, B, c_mod, C, reuse_a, reuse_b)
  return __builtin_amdgcn_wmma_f32_16x16x4_f32(false, a, false, b, (short)0, c,
                                               false, false);
}

// One radix-16 stage over the 4096-point working set.
// STAGE==1: DFT over digit n2b (stride 256), cols contiguous, twiddle w256^(t*m)
// STAGE==2: DFT over digit n2a (stride 16),  twiddle w4096^(n*(t+16m))
// STAGE==3: DFT over digit n1  (stride 1),   no twiddle, digit-shuffled store
template <int STAGE>
__device__ __forceinline__ void dft16_stage(const float* __restrict__ inR,
                                            const float* __restrict__ inI,
                                            float* __restrict__ outR,
                                            float* __restrict__ outI,
                                            const v2f* ar, const v2f* ai,
                                            const v2f* ani, int lane, int wv) {
  const int g = lane >> 4;   // half-wave group
  const int n = lane & 15;   // column within tile
#pragma unroll
  for (int half = 0; half < 2; ++half) {
    const int t = wv + 8 * half;  // tile index (16 tiles per stage, 8 waves)
    v2f br[4], bi[4];
#pragma unroll
    for (int kc = 0; kc < 4; ++kc) {
      const int r0 = 4 * kc + 2 * g;  // B rows per ISA K-striping
      int i0, i1;
      if (STAGE == 1)      { i0 = (t * 16 + n) + 256 * r0; i1 = i0 + 256; }
      else if (STAGE == 2) { i0 = n + 16 * r0 + 256 * t;   i1 = i0 + 16;  }
      else                 { i0 = r0 + 16 * n + 256 * t;   i1 = i0 + 1;   }
      br[kc][0] = inR[slotf(i0)]; br[kc][1] = inR[slotf(i1)];
      bi[kc][0] = inI[slotf(i0)]; bi[kc][1] = inI[slotf(i1)];
    }
    v8f accr = {0.f, 0.f, 0.f, 0.f, 0.f, 0.f, 0.f, 0.f};
    v8f acci = {0.f, 0.f, 0.f, 0.f, 0.f, 0.f, 0.f, 0.f};
    // Out_r = Dr*Mr + (-Di)*Mi ; Out_i = Dr*Mi + Di*Mr  (two K=16 chains each)
#pragma unroll
    for (int kc = 0; kc < 4; ++kc) {
      accr = wmma4(ar[kc], br[kc], accr);
      acci = wmma4(ar[kc], bi[kc], acci);
    }
#pragma unroll
    for (int kc = 0; kc < 4; ++kc) {
      accr = wmma4(ani[kc], bi[kc], accr);
      acci = wmma4(ai[kc],  br[kc], acci);
    }
#pragma unroll
    for (int v = 0; v < 8; ++v) {
      const int m = v + 8 * g;  // output DFT index (C/D row layout)
      float cs = 1.f, sn = 0.f;
      int oi;
      if (STAGE == 1) {
        __sincosf(-(PI2F / 256.f) * (float)(t * m), &sn, &cs);
        oi = (t * 16 + n) + 256 * m;
      } else if (STAGE == 2) {
        __sincosf(-(PI2F / 4096.f) * (float)(n * (t + 16 * m)), &sn, &cs);
        oi = n + 16 * m + 256 * t;
      } else {
        oi = t + 16 * n + 256 * m;
      }
      const float xr = accr[v], xi = acci[v];
      outR[slotf(oi)] = xr * cs - xi * sn;
      outI[slotf(oi)] = xr * sn + xi * cs;
    }
  }
}

// Forward complex FFT_4096: a -> (tmp) -> a -> out. All three stages barriered.
__device__ __forceinline__ void fft4096(float* aR, float* aI, float* tR,
                                        float* tI, float* oR, float* oI,
                                        const v2f* ar, const v2f* ai,
                                        const v2f* ani, int lane, int wv) {
  dft16_stage<1>(aR, aI, tR, tI, ar, ai, ani, lane, wv); __syncthreads();
  dft16_stage<2>(tR, tI, aR, aI, ar, ai, ani, lane, wv); __syncthreads();
  dft16_stage<3>(aR, aI, oR, oI, ar, ai, ani, lane, wv); __syncthreads();
}

__global__ __launch_bounds__(256) void fftconv_wmma_kernel(
    const float* __restrict__ x, const float* __restrict__ filt,
    float* __restrict__ y, int H, int L) {
  __shared__ float Ar_[NPAD], Ai_[NPAD];
  __shared__ float Br_[NPAD], Bi_[NPAD];
  __shared__ float Hr_[NPAD], Hi_[NPAD];
  __shared__ float Xs_[NFFT];  // TDM staging: raw x row == packed complex pairs

  const int tid  = threadIdx.x;
  const int lane = tid & 31;
  const int wv   = tid >> 5;
  const int ch   = blockIdx.x;      // b*H + h
  const int h    = ch % H;

  // --- kick off TDM DMA of the x row into LDS (overlaps with filter FFT) ---
  if (wv == 0) {
    const unsigned long long ga =
        (unsigned long long)(const void*)(x + (size_t)ch * (size_t)L);
    const unsigned int lds = (unsigned int)(unsigned long long)(void*)&Xs_[0];
    u32x4 g0;
    g0[0] = 1u;                                  // count=1 (valid user D#)
    g0[1] = lds;                                 // lds_addr
    g0[2] = (unsigned int)(ga & 0xFFFFFFFFu);    // global_addr[31:0]
    g0[3] = (unsigned int)((ga >> 32) & 0x1FFFFFFu) | (2u << 30);  // type=2
    i32x8 g1;
    g1[0] = (int)(2u << 16);                     // data_size = 4 bytes
    g1[1] = (int)((unsigned)(NFFT & 0xFFFF) << 16);  // tensor_dim0[15:0]
    g1[2] = (int)((unsigned)(NFFT >> 16) | (1u << 16));  // dim0 hi, tensor_dim1=1
    g1[3] = (int)((unsigned)NFFT << 16);         // tile_dim0 = 4096
    g1[4] = 0;                                   // tile_dim1/2 unused
    g1[5] = NFFT;                                // tensor_dim0_stride lo
    g1[6] = 0;
    g1[7] = 0;
    i32x4 z4 = {0, 0, 0, 0};
    i32x8 z8 = {0, 0, 0, 0, 0, 0, 0, 0};
    __builtin_amdgcn_tensor_load_to_lds(g0, g1, z4, z4, z8, 0);
  }

  // --- per-lane DFT16 A-operands (ISA 16x4 fp32 A layout) ---
  v2f ar[4], ai[4], ani[4];
  {
    const int m = lane & 15, g = lane >> 4;
#pragma unroll
    for (int kc = 0; kc < 4; ++kc) {
#pragma unroll
      for (int j = 0; j < 2; ++j) {
        const int k = 4 * kc + 2 * g + j;
        float s, c;
        __sincosf(-(PI2F / 16.f) * (float)(m * k), &s, &c);
        ar[kc][j] = c; ai[kc][j] = s; ani[kc][j] = -s;
      }
    }
  }

  // --- 1) filter spectrum: pack real 4096 (+zero pad) as 2048 complex ---
  const float* frow = filt + (size_t)h * (size_t)L;
#pragma unroll
  for (int j = 0; j < 16; ++j) {
    const int i = tid + 256 * j;
    float re = 0.f, im = 0.f;
    if (i < 2048) { const float2 v = ((const float2*)frow)[i]; re = v.x; im = v.y; }
    Ar_[slotf(i)] = re; Ai_[slotf(i)] = im;
  }
  __syncthreads();
  fft4096(Ar_, Ai_, Br_, Bi_, Hr_, Hi_, ar, ai, ani, lane, wv);  // Zh -> H

  // --- 2) x spectrum: pack from TDM-staged row (z[k] = x[2k] + i*x[2k+1]) ---
  if (wv == 0) { __builtin_amdgcn_s_wait_tensorcnt(0); }
  __syncthreads();
#pragma unroll
  for (int j = 0; j < 16; ++j) {
    const int i = tid + 256 * j;
    float re = 0.f, im = 0.f;
    if (i < 2048) { re = Xs_[2 * i]; im = Xs_[2 * i + 1]; }
    Ar_[slotf(i)] = re; Ai_[slotf(i)] = im;
  }
  __syncthreads();
  fft4096(Ar_, Ai_, Br_, Bi_, Br_, Bi_, ar, ai, ani, lane, wv);  // Z -> B

  // --- 3) fused unpack (rfft_8192) * multiply * repack(conj) -> V in A ---
  const float sc = 1.0f / 4096.0f;  // IFFT 1/N folded in
#pragma unroll 1
  for (int j = 0; j < 8; ++j) {
    const int k = tid + 256 * j;
    if (k == 0) {
      // bins 0 and 2048 (self-paired, real / conj cases)
      const float z0r = Br_[slotf(0)], z0i = Bi_[slotf(0)];
      const float h0r = Hr_[slotf(0)], h0i = Hi_[slotf(0)];
      const float X0 = z0r + z0i, XN = z0r - z0i;  // X[0], X[4096] (real)
      const float G0 = h0r + h0i, GN = h0r - h0i;
      const float Y0 = X0 * G0, YN = XN * GN;
      Ar_[slotf(0)] = sc * 0.5f * (Y0 + YN);
      Ai_[slotf(0)] = -sc * 0.5f * (Y0 - YN);
      const float zr2 = Br_[slotf(2048)], zi2 = Bi_[slotf(2048)];
      const float hr2 = Hr_[slotf(2048)], hi2 = Hi_[slotf(2048)];
      Ar_[slotf(2048)] = sc * (zr2 * hr2 - zi2 * hi2);
      Ai_[slotf(2048)] = -sc * (zr2 * hi2 + zi2 * hr2);
    } else {
      const int kN = NFFT - k;
      const float zer = Br_[slotf(k)],  zei = Bi_[slotf(k)];
      const float znr = Br_[slotf(kN)], zni = Bi_[slotf(kN)];
      const float her = Hr_[slotf(k)],  hei = Hi_[slotf(k)];
      const float hnr = Hr_[slotf(kN)], hni = Hi_[slotf(kN)];
      float st, ct;
      __sincosf(-(PI2F / 8192.f) * (float)k, &st, &ct);  // t = e^{-2pi i k/8192}
      // even/odd split of x-spectrum
      const float xer = 0.5f * (zer + znr), xei = 0.5f * (zei - zni);
      const float xo_r = 0.5f * (zei + zni), xo_i = -0.5f * (zer - znr);
      // even/odd split of filter spectrum
      const float ger = 0.5f * (her + hnr), gei = 0.5f * (hei - hni);
      const float go_r = 0.5f * (hei + hni), go_i = -0.5f * (her - hnr);
      // t * odd
      const float txr = ct * xo_r - st * xo_i, txi = ct * xo_i + st * xo_r;
      const float tgr = ct * go_r - st * go_i, tgi = ct * go_i + st * go_r;
      // X[k], conj(X[N-k]); same for H
      const float x1r = xer + txr, x1i = xei + txi;
      const float x2r = xer - txr, x2i = xei - txi;
      const float h1r = ger + tgr, h1i = gei + tgi;
      const float h2r = ger - tgr, h2i = gei - tgi;
      // Y[k] = X*H ; Y2c with Y[N-k] = conj(Y2c)
      const float y1r = x1r * h1r - x1i * h1i, y1i = x1r * h1i + x1i * h1r;
      const float y2r = x2r * h2r - x2i * h2i, y2i = x2r * h2i + x2i * h2r;
      // repack: W[k] = A + Bq, W[N-k] = conj(A - Bq); store V = sc*conj(W)
      const float aqr = 0.5f * (y1r + y2r), aqi = 0.5f * (y1i + y2i);
      const float ur = y1r - y2r, ui = y1i - y2i;
      const float bqr = 0.5f * (st * ur - ct * ui);
      const float bqi = 0.5f * (st * ui + ct * ur);
      Ar_[slotf(k)]  = sc * (aqr + bqr);
      Ai_[slotf(k)]  = -sc * (aqi + bqi);
      Ar_[slotf(kN)] = sc * (aqr - bqr);
      Ai_[slotf(kN)] = sc * (aqi - bqi);
    }
  }
  __syncthreads();

  // --- 4) inverse FFT via conj trick: R = FFT(V); y pairs = conj(R) ---
  fft4096(Ar_, Ai_, Br_, Bi_, Br_, Bi_, ar, ai, ani, lane, wv);  // R -> B

  float2* yrow = (float2*)(y + (size_t)ch * (size_t)L);
#pragma unroll
  for (int j = 0; j < 8; ++j) {
    const int i = tid + 256 * j;  // i in [0,2048): outputs y[2i], y[2i+1]
    float2 o;
    o.x = Br_[slotf(i)];
    o.y = -Bi_[slotf(i)];
    yrow[i] = o;
  }
}

extern "C" void kernel_launch(void* const* d_in, const int* in_sizes, int n_in,
                              void* d_out, int out_size, void* d_ws,
                              size_t ws_size, hipStream_t stream) {
  (void)in_sizes; (void)n_in; (void)d_ws; (void)ws_size; (void)out_size;
  const float* x    = (const float*)d_in[0];
  const float* filt = (const float*)d_in[1];
  float* y          = (float*)d_out;
  const int B = 8, H = 768, L = 4096;
  fftconv_wmma_kernel<<<dim3(B * H), dim3(256), 0, stream>>>(x, filt, y, H, L);
}